// NN_Pooling_63410897158182
// MI455X (gfx1250) — compile-verified
//
#include <hip/hip_runtime.h>

// CDNA5 / gfx1250: wave32, WMMA f32 16x16x4.
typedef __attribute__((ext_vector_type(2))) float v2f;
typedef __attribute__((ext_vector_type(8))) float v8f;

#define N_TRACKS 4096
#define NTILES   (N_TRACKS / 16)
#define NTOP 4
#define EMB 8
#define BLOCK_THREADS 256
#define QUERIES_PER_BLOCK 128   // 8 waves * 16 queries
#define BIGF 3.0e38f

__global__ __launch_bounds__(BLOCK_THREADS)
void knn_pool_wmma_kernel(const float* __restrict__ obs1,
                          const float* __restrict__ obs2,
                          const float* __restrict__ W,      // [4,8] row-major (in,out)
                          const float* __restrict__ bias,   // [8]
                          float* __restrict__ out)          // [4096,32]
{
    // Per-point GEMM feature: (x, y, |p|^2, 1). 4096 * 16B = 64KB LDS (of 320KB/WGP).
    __shared__ float4 sF[N_TRACKS];

    const int tid = threadIdx.x;
    for (int j = tid; j < N_TRACKS; j += BLOCK_THREADS) {
        float x = obs2[2 * j + 0];
        float y = obs2[2 * j + 1];
        sF[j] = make_float4(x, y, x * x + y * y, 1.0f);
    }
    __syncthreads();

    const int lane = tid & 31;
    const int wave = tid >> 5;
    const int col  = lane & 15;          // query column / A-row within tile
    const bool hi  = lane >= 16;         // upper half-wave
    const int q    = blockIdx.x * QUERIES_PER_BLOCK + wave * 16 + col;

    // B operand (4x16, K x N), column q: [-2x, -2y, 1, |q|^2].
    // CDNA5 f32 B layout: lanes 0-15 carry K=0 (v0), K=1 (v1); lanes 16-31 carry K=2, K=3.
    const float4 qf = sF[q];
    v2f bmat;
    bmat.x = hi ? 1.0f : (-2.0f * qf.x);
    bmat.y = hi ? qf.z : (-2.0f * qf.y);

    // Running top-4 (ascending squared distance), matching top_k order.
    float b0 = BIGF, b1 = BIGF, b2 = BIGF, b3 = BIGF;
    int   i0 = 0,    i1 = 0,    i2 = 0,    i3 = 0;

    auto insert = [&](float d, int j) {
        if (d < b3) {
            bool c0 = d < b0, c1 = d < b1, c2 = d < b2;
            float nb3 = c2 ? b2 : d;              int ni3 = c2 ? i2 : j;
            float nb2 = c2 ? (c1 ? b1 : d) : b2;  int ni2 = c2 ? (c1 ? i1 : j) : i2;
            float nb1 = c1 ? (c0 ? b0 : d) : b1;  int ni1 = c1 ? (c0 ? i0 : j) : i1;
            float nb0 = c0 ? d : b0;              int ni0 = c0 ? j : i0;
            b0 = nb0; b1 = nb1; b2 = nb2; b3 = nb3;
            i0 = ni0; i1 = ni1; i2 = ni2; i3 = ni3;
        }
    };

    // Sweep all candidate tiles: one ds_load + one v_wmma_f32_16x16x4_f32 each,
    // software-pipelined so the next tile's LDS load overlaps this tile's math.
    float4 cf = sF[col];                 // tile 0 candidate for this lane
    for (int t = 0; t < NTILES; ++t) {
        const float4 nf = sF[(((t + 1) & (NTILES - 1)) << 4) + col];  // prefetch next

        // A operand (16x4, M x K), row = candidate t*16+col: [x, y, |c|^2, 1].
        // f32 A layout: lanes 0-15 carry K=0,1; lanes 16-31 carry K=2,3.
        v2f amat;
        amat.x = hi ? cf.z : cf.x;
        amat.y = hi ? 1.0f : cf.y;

        v8f acc = {};
        // D[m][n] = |cand_m - query_n|^2 in one WMMA.
        acc = __builtin_amdgcn_wmma_f32_16x16x4_f32(
                  false, amat, false, bmat, (short)0, acc, false, false);

        // Whole-tile rejection: min-reduce the 8 distances in this lane's column.
        // Self-distance (~0) guarantees entry on the one tile containing q, where
        // the per-value self mask below kicks it out.
        float m01 = fminf(acc[0], acc[1]);
        float m23 = fminf(acc[2], acc[3]);
        float m45 = fminf(acc[4], acc[5]);
        float m67 = fminf(acc[6], acc[7]);
        float tmin = fminf(fminf(m01, m23), fminf(m45, m67));

        if (tmin < b3) {                 // rare after warm-up -> execz skip
            const int jbase = (t << 4) + (hi ? 8 : 0);
#pragma unroll
            for (int p = 0; p < 8; ++p) {
                float d = acc[p];
                int   j = jbase + p;
                d = (j == q) ? BIGF : d; // exclude self (off-diagonal in reference)
                insert(d, j);
            }
        }
        cf = nf;
    }

    // Query column q is split across lane pair (lane, lane^16): merge their top-4s.
    {
        float ob0 = __shfl_xor(b0, 16, 32), ob1 = __shfl_xor(b1, 16, 32);
        float ob2 = __shfl_xor(b2, 16, 32), ob3 = __shfl_xor(b3, 16, 32);
        int   oi0 = __shfl_xor(i0, 16, 32), oi1 = __shfl_xor(i1, 16, 32);
        int   oi2 = __shfl_xor(i2, 16, 32), oi3 = __shfl_xor(i3, 16, 32);
        insert(ob0, oi0);
        insert(ob1, oi1);
        insert(ob2, oi2);
        insert(ob3, oi3);
    }

    // Fused Linear(4 -> 8) + ReLU on the 4 nearest neighbors; lanes 0-15 own the queries.
    if (!hi) {
        const int nn[NTOP] = { i0, i1, i2, i3 };
        const float qx = qf.x, qy = qf.y;
        const float qvx = qx - obs1[2 * q + 0];
        const float qvy = qy - obs1[2 * q + 1];
        float* o = out + q * (NTOP * EMB);
#pragma unroll
        for (int k = 0; k < NTOP; ++k) {
            const int j = nn[k];
            const float jx = sF[j].x, jy = sF[j].y;
            const float jvx = jx - obs1[2 * j + 0];
            const float jvy = jy - obs1[2 * j + 1];
            const float f0 = jx - qx;            // rel_pos.x
            const float f1 = jy - qy;            // rel_pos.y
            const float f2 = jvx - qvx;          // rel_vel.x
            const float f3 = jvy - qvy;          // rel_vel.y
#pragma unroll
            for (int e = 0; e < EMB; ++e) {
                float a = bias[e]
                        + f0 * W[0 * EMB + e]
                        + f1 * W[1 * EMB + e]
                        + f2 * W[2 * EMB + e]
                        + f3 * W[3 * EMB + e];
                o[k * EMB + e] = fmaxf(a, 0.0f);
            }
        }
    }
}

extern "C" void kernel_launch(void* const* d_in, const int* in_sizes, int n_in,
                              void* d_out, int out_size, void* d_ws, size_t ws_size,
                              hipStream_t stream) {
    const float* obs1 = (const float*)d_in[0];   // [4096,2]
    const float* obs2 = (const float*)d_in[1];   // [4096,2]
    const float* W    = (const float*)d_in[2];   // [4,8]
    const float* b    = (const float*)d_in[3];   // [8]
    float* out        = (float*)d_out;           // [4096,32]

    knn_pool_wmma_kernel<<<N_TRACKS / QUERIES_PER_BLOCK, BLOCK_THREADS, 0, stream>>>(
        obs1, obs2, W, b, out);
}